// TemporalAttention_47304769798690
// MI455X (gfx1250) — compile-verified
//
#include <hip/hip_runtime.h>
#include <math.h>

typedef __attribute__((ext_vector_type(2))) float v2f;
typedef __attribute__((ext_vector_type(8))) float v8f;

#define B_    4
#define TP_   16
#define S_    256
#define D_    2048
#define T_    64      // total tokens = 16+16+16+16
#define H_    16
#define HD_   128
#define NQKV_ 6144
#define KB_   32      // K-chunk staged in LDS per GEMM stage

// ---------------------------------------------------------------------------
// 1) tokens = [mean_S(v_p), mean_S(v_f), a_p, a_f]  -> (B, 64, 2048)
// ---------------------------------------------------------------------------
__global__ void ta_tokens_kernel(const float* __restrict__ vp,
                                 const float* __restrict__ vf,
                                 const float* __restrict__ ap,
                                 const float* __restrict__ af,
                                 float4* __restrict__ tok4) {
  int i = blockIdx.x * blockDim.x + threadIdx.x;     // over B*T*(D/4) = 131072
  if (i >= B_ * T_ * (D_ / 4)) return;
  int c4 = i & (D_ / 4 - 1);                         // 0..511
  int bt = i >> 9;
  int b = bt >> 6, t = bt & 63;
  float4 r;
  if (t < 32) {
    const float* src = (t < 16) ? vp : vf;
    int tt = t & 15;
    const float4* p = (const float4*)src +
                      (size_t)(b * TP_ + tt) * S_ * (D_ / 4) + c4;
    float acx = 0.f, acy = 0.f, acz = 0.f, acw = 0.f;
    for (int s = 0; s < S_; ++s) {
      float4 x = p[(size_t)s * (D_ / 4)];
      acx += x.x; acy += x.y; acz += x.z; acw += x.w;
    }
    const float inv = 1.0f / (float)S_;
    r = make_float4(acx * inv, acy * inv, acz * inv, acw * inv);
  } else {
    const float* src = (t < 48) ? ap : af;
    int tt = t & 15;
    r = ((const float4*)src)[(size_t)(b * TP_ + tt) * (D_ / 4) + c4];
  }
  tok4[i] = r;
}

// ---------------------------------------------------------------------------
// 2/4) fp32 WMMA GEMM: C[MxN] = A[MxK] @ B[KxN] (+ bias)
//   Block: 256 threads = 8 waves; strip = 16 rows x 128 cols, 16x16 per wave.
//   K staged in LDS chunks of 32, pair-interleaved so every WMMA fragment is
//   one aligned ds_load_b64. Global loads of the next stage are prefetched
//   into registers while the current 8-wmma chain executes.
// ---------------------------------------------------------------------------
__global__ void __launch_bounds__(256)
ta_wmma_gemm_kernel(const float* __restrict__ A,
                    const float* __restrict__ Bm,
                    const float* __restrict__ bias,
                    float* __restrict__ C,
                    int M, int N, int K) {
  // LDS, pair-interleaved: element (k, x) lives at ((k>>1)*X + x)*2 + (k&1)
  __shared__ float As2[16 * KB_];        // 512 floats (2 KB)
  __shared__ float Bs2[KB_ * 128];       // 4096 floats (16 KB)

  const int tid  = threadIdx.x;
  const int wave = tid >> 5;
  const int lane = tid & 31;
  const int half = lane >> 4;            // 0: K pair {0,1}; 1: K pair {2,3}
  const int l16  = lane & 15;
  const int m0 = blockIdx.y * 16;
  const int n0 = blockIdx.x * 128;
  const int wn = wave * 16;              // wave's column offset in the strip

  // ---- staging index precompute ------------------------------------------
  // A: 128 float4 (16 rows x 8 float4-of-K); threads 0..127
  const bool hasA = tid < 128;
  const int am = tid >> 3;               // row within tile (0..15)
  const int ak = (tid & 7) * 4;          // k within chunk (0,4,...,28)
  const float4* agp =
      hasA ? (const float4*)(A + (size_t)(m0 + am) * K + ak) : (const float4*)A;
  // B: 1024 float4 (32 k-rows x 32 float4 cols); 4 per thread
  const int bk0 = tid >> 5;              // first k-row (0..7), +8 per rep
  const int bc  = (tid & 31) * 4;        // col within 128

  float4 aReg = {};
  float4 bReg[4];
  // prologue: fetch stage 0
  if (hasA) aReg = agp[0];
#pragma unroll
  for (int r = 0; r < 4; ++r) {
    const int k = bk0 + 8 * r;
    bReg[r] = *(const float4*)(Bm + (size_t)k * N + n0 + bc);
  }

  v8f acc = {};
  for (int ks = 0; ks < K; ks += KB_) {
    __syncthreads();
    // ---- store staged registers to LDS (pair-interleaved) ----------------
    if (hasA) {
      const int p = ((ak >> 1) * 16 + am) * 2;         // k even pair base
      *(float2*)&As2[p]      = make_float2(aReg.x, aReg.y);
      *(float2*)&As2[p + 32] = make_float2(aReg.z, aReg.w);   // k+2 pair
    }
#pragma unroll
    for (int r = 0; r < 4; ++r) {
      const int k = bk0 + 8 * r;
      const int base = ((k >> 1) * 128 + bc) * 2 + (k & 1);
      Bs2[base]     = bReg[r].x;
      Bs2[base + 2] = bReg[r].y;
      Bs2[base + 4] = bReg[r].z;
      Bs2[base + 6] = bReg[r].w;
    }
    __syncthreads();

    // ---- prefetch next stage into registers while wmma chain runs --------
    const int ksn = ks + KB_;
    if (ksn < K) {
      if (hasA) aReg = agp[ksn >> 2];
#pragma unroll
      for (int r = 0; r < 4; ++r) {
        const int k = ksn + bk0 + 8 * r;
        bReg[r] = *(const float4*)(Bm + (size_t)k * N + n0 + bc);
      }
      __builtin_prefetch(Bm + (size_t)(ksn + KB_) * N + n0 + bc, 0, 3);
    }

    // ---- 8 chained WMMAs from LDS ----------------------------------------
#pragma unroll
    for (int kk = 0; kk < KB_; kk += 4) {
      const v2f a = *(const v2f*)&As2[(((kk >> 1) + half) * 16 + l16) * 2];
      const v2f b = *(const v2f*)&Bs2[(((kk >> 1) + half) * 128 + wn + l16) * 2];
      acc = __builtin_amdgcn_wmma_f32_16x16x4_f32(
          /*neg_a=*/false, a, /*neg_b=*/false, b,
          /*c_mod=*/(short)0, acc, /*reuse_a=*/false, /*reuse_b=*/false);
    }
  }

  // D layout: VGPR r -> row m0 + 8*half + r, col n0 + wn + l16
  const float bv = bias ? bias[n0 + wn + l16] : 0.0f;
#pragma unroll
  for (int r = 0; r < 8; ++r)
    C[(size_t)(m0 + 8 * half + r) * N + n0 + wn + l16] = acc[r] + bv;
}

// ---------------------------------------------------------------------------
// 3) fused RoPE + causal softmax attention, one (b, h) per block (T=64,hd=128)
// ---------------------------------------------------------------------------
__global__ void ta_attention_kernel(const float* __restrict__ qkv,
                                    float* __restrict__ attn_out) {
  extern __shared__ float smem[];
  float* qs = smem;                 // 64*128
  float* ks = smem + T_ * HD_;      // 64*128
  float* vs = smem + 2 * T_ * HD_;  // 64*128
  float* sc = smem + 3 * T_ * HD_;  // 64*64

  const int tid = threadIdx.x;                 // 256 threads
  const int b = blockIdx.x >> 4;
  const int h = blockIdx.x & 15;
  const size_t qoff = (size_t)b * T_ * NQKV_ + (size_t)h * HD_;

  // Stage 1: load + RoPE q,k; load v
  for (int idx = tid; idx < T_ * HD_; idx += blockDim.x) {
    const int t = idx >> 7;
    const int d = idx & 127;
    const int fi = d & 63;
    // inv_freq = 10000^{-2*fi/128} = exp(-fi * ln(10000)/64)
    const float ang = (float)t * __expf(-0.14391156f * (float)fi);
    const float cc = __cosf(ang), ss = __sinf(ang);
    const size_t base = qoff + (size_t)t * NQKV_ + d;
    const float qv = qkv[base];
    const float kv = qkv[base + D_];
    const size_t pair = qoff + (size_t)t * NQKV_ + ((d < 64) ? (d + 64) : (d - 64));
    const float qr = (d < 64) ? -qkv[pair] : qkv[pair];
    const float kr = (d < 64) ? -qkv[pair + D_] : qkv[pair + D_];
    qs[idx] = qv * cc + qr * ss;
    ks[idx] = kv * cc + kr * ss;
    vs[idx] = qkv[base + 2 * D_];
  }
  __syncthreads();

  // Stage 2: scores (causal: only tk <= tq)
  const float scale = 0.08838834764831845f;    // 128^-0.5
  for (int p = tid; p < T_ * T_; p += blockDim.x) {
    const int tq = p >> 6, tk = p & 63;
    float s = 0.f;
    if (tk <= tq) {
      const float* qr = qs + tq * HD_;
      const float* kr = ks + tk * HD_;
      for (int d = 0; d < HD_; ++d) s += qr[d] * kr[d];
      s *= scale;
    }
    sc[p] = s;
  }
  __syncthreads();

  // Stage 3: softmax per row (one thread per row)
  if (tid < T_) {
    const int tq = tid;
    float m = -1e30f;
    for (int tk = 0; tk <= tq; ++tk) m = fmaxf(m, sc[tq * 64 + tk]);
    float sum = 0.f;
    for (int tk = 0; tk <= tq; ++tk) {
      const float e = __expf(sc[tq * 64 + tk] - m);
      sc[tq * 64 + tk] = e;
      sum += e;
    }
    const float inv = 1.0f / sum;
    for (int tk = 0; tk <= tq; ++tk) sc[tq * 64 + tk] *= inv;
    for (int tk = tq + 1; tk < 64; ++tk) sc[tq * 64 + tk] = 0.f;
  }
  __syncthreads();

  // Stage 4: out = attn @ v -> attn_out (B, T, D) at col h*128+d
  for (int idx = tid; idx < T_ * HD_; idx += blockDim.x) {
    const int t = idx >> 7;
    const int d = idx & 127;
    float acc = 0.f;
    const float* srow = sc + t * 64;
    for (int tk = 0; tk < T_; ++tk) acc += srow[tk] * vs[tk * HD_ + d];
    attn_out[((size_t)(b * T_ + t)) * D_ + h * HD_ + d] = acc;
  }
}

// ---------------------------------------------------------------------------
// 5) scatter/broadcast outputs: [v_p_out | v_f_out | a_p_out | a_f_out]
// ---------------------------------------------------------------------------
__global__ void ta_scatter_kernel(const float4* __restrict__ p4,
                                  float4* __restrict__ out4) {
  const long D4  = D_ / 4;                       // 512
  const long VP4 = (long)B_ * TP_ * S_ * D4;     // 8388608
  const long A4  = (long)B_ * TP_ * D4;          // 32768
  const long total = 2 * VP4 + 2 * A4;
  const long stride = (long)gridDim.x * blockDim.x;
  for (long i = (long)blockIdx.x * blockDim.x + threadIdx.x; i < total; i += stride) {
    long token, c4;
    if (i < 2 * VP4) {
      const long j = (i < VP4) ? i : i - VP4;
      c4 = j & (D4 - 1);
      const long row = j >> 9;                   // (b*16+t)*256 + s
      const long bt = row >> 8;
      const long b = bt >> 4, t = bt & 15;
      token = b * T_ + t + ((i < VP4) ? 0 : TP_);
    } else {
      const long j = i - 2 * VP4;
      const long ja = (j < A4) ? j : j - A4;
      c4 = ja & (D4 - 1);
      const long bt = ja >> 9;
      const long b = bt >> 4, t = bt & 15;
      token = b * T_ + t + ((j < A4) ? 2 * TP_ : 3 * TP_);
    }
    out4[i] = p4[token * D4 + c4];
  }
}

// ---------------------------------------------------------------------------
extern "C" void kernel_launch(void* const* d_in, const int* in_sizes, int n_in,
                              void* d_out, int out_size, void* d_ws, size_t ws_size,
                              hipStream_t stream) {
  const float* v_p    = (const float*)d_in[0];
  const float* v_f    = (const float*)d_in[1];
  const float* a_p    = (const float*)d_in[2];
  const float* a_f    = (const float*)d_in[3];
  const float* W_qkv  = (const float*)d_in[4];
  const float* W_proj = (const float*)d_in[5];
  const float* b_proj = (const float*)d_in[6];

  float* ws = (float*)d_ws;
  float* tokens = ws;                                   // 256*2048  = 524288
  float* qkv    = tokens + (size_t)B_ * T_ * D_;        // 256*6144  = 1572864
  float* attn   = qkv + (size_t)B_ * T_ * NQKV_;        // 256*2048
  float* proj   = attn + (size_t)B_ * T_ * D_;          // 256*2048

  // 1) tokens
  {
    const int n = B_ * T_ * (D_ / 4);                   // 131072
    ta_tokens_kernel<<<(n + 255) / 256, 256, 0, stream>>>(
        v_p, v_f, a_p, a_f, (float4*)tokens);
  }
  // 2) qkv = tokens @ W_qkv   (M=256, N=6144, K=2048)
  {
    dim3 grid(NQKV_ / 128, (B_ * T_) / 16);             // (48, 16)
    ta_wmma_gemm_kernel<<<grid, 256, 0, stream>>>(
        tokens, W_qkv, nullptr, qkv, B_ * T_, NQKV_, D_);
  }
  // 3) rope + attention  (64 blocks = B*H)
  {
    const size_t lds = (size_t)(3 * T_ * HD_ + T_ * T_) * sizeof(float); // 114688
    ta_attention_kernel<<<B_ * H_, 256, lds, stream>>>(qkv, attn);
  }
  // 4) proj = attn @ W_proj + b_proj  (M=256, N=2048, K=2048)
  {
    dim3 grid(D_ / 128, (B_ * T_) / 16);                // (16, 16)
    ta_wmma_gemm_kernel<<<grid, 256, 0, stream>>>(
        attn, W_proj, b_proj, proj, B_ * T_, D_, D_);
  }
  // 5) broadcast/scatter outputs
  {
    ta_scatter_kernel<<<4096, 256, 0, stream>>>((const float4*)proj,
                                                (float4*)d_out);
  }
}